// signedGCN_11227044512441
// MI455X (gfx1250) — compile-verified
//
#include <hip/hip_runtime.h>

typedef __attribute__((ext_vector_type(2))) float v2f;
typedef __attribute__((ext_vector_type(8))) float v8f;

#define N_NODES   100000
#define N_EDGES   1600000
#define IN_C      32
#define HID_C     64
#define OUT_C     32
#define N_GRAPHS  64
#define NEG_SLOPE 0.01f

// ---------------------------------------------------------------------------
// deg[row[e]] += |ew[e]|   (signed-GCN degree from |edge_weight| at source)
// ---------------------------------------------------------------------------
__global__ void deg_kernel(const int* __restrict__ row, const float* __restrict__ ew,
                           float* __restrict__ deg) {
    int e = blockIdx.x * blockDim.x + threadIdx.x;
    if (e < N_EDGES) unsafeAtomicAdd(&deg[row[e]], fabsf(ew[e]));
}

// dis[i] = deg>0 ? deg^-0.5 : 0
__global__ void dis_kernel(const float* __restrict__ deg, float* __restrict__ dis) {
    int i = blockIdx.x * blockDim.x + threadIdx.x;
    if (i < N_NODES) {
        float d = deg[i];
        dis[i] = (d > 0.0f) ? rsqrtf(d) : 0.0f;
    }
}

// ---------------------------------------------------------------------------
// H[M,N] = X[M,K] @ W[K,N] via V_WMMA_F32_16X16X4_F32 (full f32 matrix pipe).
// One wave owns a 16-row stripe; 100000 rows / 16 = 6250 waves exactly, so
// EXEC is all-1s everywhere (WMMA requirement). A-frag layout (ISA 16x4 f32):
// lanes 0-15 hold (row, K=k0/k0+1), lanes 16-31 hold (row, K=k0+2/k0+3).
// B-frag mirrors (VGPR0 = rows k0 / k0+2, VGPR1 = rows k0+1 / k0+3).
// C/D: 8 VGPRs, VGPR r -> (M = r + 8*half, N = lane%16).
// ---------------------------------------------------------------------------
template<int K, int N>
__global__ void gemm_wmma_kernel(const float* __restrict__ X,
                                 const float* __restrict__ W,
                                 float* __restrict__ H) {
    const int wave = (blockIdx.x * blockDim.x + threadIdx.x) >> 5;
    const int lane = threadIdx.x & 31;
    const int half = lane >> 4;   // 0: K pair {0,1}; 1: K pair {2,3}
    const int l16  = lane & 15;
    const int row0 = wave * 16;

    const float* __restrict__ xrow = X + (size_t)(row0 + l16) * K + 2 * half;

#pragma unroll
    for (int n0 = 0; n0 < N; n0 += 16) {
        v8f acc = {};
#pragma unroll
        for (int k = 0; k < K; k += 4) {
            v2f a, b;
            a.x = xrow[k];
            a.y = xrow[k + 1];
            const float* wp = W + (size_t)(k + 2 * half) * N + n0 + l16;
            b.x = wp[0];
            b.y = wp[N];
            acc = __builtin_amdgcn_wmma_f32_16x16x4_f32(
                /*neg_a=*/false, a, /*neg_b=*/false, b,
                /*c_mod=*/(short)0, acc, /*reuse_a=*/false, /*reuse_b=*/false);
        }
        float* hp = H + (size_t)(row0 + 8 * half) * N + n0 + l16;
#pragma unroll
        for (int r = 0; r < 8; ++r) hp[(size_t)r * N] = acc[r];
    }
}

// ---------------------------------------------------------------------------
// Edge scatter: OUT[col] += dis[row]*dis[col] * H[row]   (C channels)
// C/16 threads per edge: each thread loads edge metadata ONCE and covers 16
// channels as 4 float4 gathers + 16 non-returning global_atomic_add_f32.
// Cuts the redundant row/col/dis metadata traffic 4x vs. C/4 threads/edge
// while keeping 6.4M (layer1) / 3.2M (layer2) threads of parallelism.
// ---------------------------------------------------------------------------
template<int C>
__global__ void scatter_kernel(const int* __restrict__ rows, const int* __restrict__ cols,
                               const float* __restrict__ dis, const float* __restrict__ H,
                               float* __restrict__ OUT) {
    const int TPE = C / 16;  // threads per edge
    long long gid = (long long)blockIdx.x * blockDim.x + threadIdx.x;
    long long e = gid / TPE;
    if (e >= N_EDGES) return;
    int cg = (int)(gid % TPE) * 16;

    int r = rows[e];
    int c = cols[e];
    float nrm = dis[r] * dis[c];

    const float* hp = H + (size_t)r * C + cg;
    float* op = OUT + (size_t)c * C + cg;
#pragma unroll
    for (int q = 0; q < 16; q += 4) {
        const float4 v = *(const float4*)(hp + q);
        unsafeAtomicAdd(op + q + 0, nrm * v.x);
        unsafeAtomicAdd(op + q + 1, nrm * v.y);
        unsafeAtomicAdd(op + q + 2, nrm * v.z);
        unsafeAtomicAdd(op + q + 3, nrm * v.w);
    }
}

// A[i] = leaky_relu(A[i] + b[i % C])
template<int C>
__global__ void bias_lrelu_kernel(float* __restrict__ A, const float* __restrict__ b, int total) {
    int i = blockIdx.x * blockDim.x + threadIdx.x;
    if (i < total) {
        float y = A[i] + b[i & (C - 1)];
        A[i] = (y > 0.0f) ? y : NEG_SLOPE * y;
    }
}

// ---------------------------------------------------------------------------
// Mean-pool stage 1: per-block LDS accumulation (ds_add_f32), then flush to
// global sums/counts with a handful of global atomics per block.
// ---------------------------------------------------------------------------
__global__ void pool_kernel(const float* __restrict__ A, const int* __restrict__ batch,
                            float* __restrict__ sums, float* __restrict__ counts) {
    __shared__ float ls[N_GRAPHS * OUT_C];   // 8 KB
    __shared__ float lc[N_GRAPHS];
    for (int i = threadIdx.x; i < N_GRAPHS * OUT_C; i += blockDim.x) ls[i] = 0.0f;
    for (int i = threadIdx.x; i < N_GRAPHS; i += blockDim.x) lc[i] = 0.0f;
    __syncthreads();

    const int NB = 512;  // nodes per block
    long long base = (long long)blockIdx.x * NB;
    for (int j = threadIdx.x; j < NB * OUT_C; j += blockDim.x) {
        long long n = base + j / OUT_C;
        int c = j & (OUT_C - 1);
        if (n < N_NODES) {
            int g = batch[n];
            unsafeAtomicAdd(&ls[g * OUT_C + c], A[n * OUT_C + c]);
            if (c == 0) unsafeAtomicAdd(&lc[g], 1.0f);
        }
    }
    __syncthreads();

    for (int i = threadIdx.x; i < N_GRAPHS * OUT_C; i += blockDim.x)
        unsafeAtomicAdd(&sums[i], ls[i]);
    for (int i = threadIdx.x; i < N_GRAPHS; i += blockDim.x)
        unsafeAtomicAdd(&counts[i], lc[i]);
}

// out[g, :2] = (sums[g]/max(count,1)) @ Wm + bm
__global__ void head_kernel(const float* __restrict__ sums, const float* __restrict__ counts,
                            const float* __restrict__ Wm, const float* __restrict__ bm,
                            float* __restrict__ out) {
    int g = threadIdx.x;
    if (g < N_GRAPHS) {
        float inv = 1.0f / fmaxf(counts[g], 1.0f);
        float acc0 = bm[0], acc1 = bm[1];
#pragma unroll
        for (int c = 0; c < OUT_C; ++c) {
            float p = sums[g * OUT_C + c] * inv;
            acc0 += p * Wm[c * 2 + 0];
            acc1 += p * Wm[c * 2 + 1];
        }
        out[g * 2 + 0] = acc0;
        out[g * 2 + 1] = acc1;
    }
}

// ---------------------------------------------------------------------------
extern "C" void kernel_launch(void* const* d_in, const int* in_sizes, int n_in,
                              void* d_out, int out_size, void* d_ws, size_t ws_size,
                              hipStream_t stream) {
    const float* x    = (const float*)d_in[0];
    const int*   ei   = (const int*)d_in[1];     // [2, E] flat
    const float* ew   = (const float*)d_in[2];
    const int*   bat  = (const int*)d_in[3];
    const float* W1   = (const float*)d_in[4];
    const float* b1   = (const float*)d_in[5];
    const float* W2   = (const float*)d_in[6];
    const float* b2   = (const float*)d_in[7];
    const float* Wm   = (const float*)d_in[8];
    const float* bm   = (const float*)d_in[9];
    float* out = (float*)d_out;

    const int* rows = ei;
    const int* cols = ei + N_EDGES;

    // Workspace layout (floats). Zeroed region first (atomic accumulators).
    float* ws = (float*)d_ws;
    size_t off = 0;
    float* deg    = ws + off; off += N_NODES;
    float* a1     = ws + off; off += (size_t)N_NODES * HID_C;   // conv1 scatter accum
    float* a2     = ws + off; off += (size_t)N_NODES * OUT_C;   // conv2 scatter accum
    float* sums   = ws + off; off += N_GRAPHS * OUT_C;
    float* counts = ws + off; off += N_GRAPHS;
    size_t zero_floats = off;                                   // memset to here
    float* dis    = ws + off; off += N_NODES;
    float* h1     = ws + off; off += (size_t)N_NODES * HID_C;   // x @ W1
    float* h2     = ws + off; off += (size_t)N_NODES * OUT_C;   // a1 @ W2
    (void)ws_size; (void)n_in; (void)in_sizes; (void)out_size;

    hipMemsetAsync(d_ws, 0, zero_floats * sizeof(float), stream);

    // degree + normalization
    deg_kernel<<<(N_EDGES + 255) / 256, 256, 0, stream>>>(rows, ew, deg);
    dis_kernel<<<(N_NODES + 255) / 256, 256, 0, stream>>>(deg, dis);

    // layer 1: h1 = x @ W1 ; scatter ; bias + leaky-relu
    gemm_wmma_kernel<IN_C, HID_C><<<N_NODES / 32, 64, 0, stream>>>(x, W1, h1);
    {
        long long total = (long long)N_EDGES * (HID_C / 16);
        scatter_kernel<HID_C><<<(int)((total + 255) / 256), 256, 0, stream>>>(rows, cols, dis, h1, a1);
    }
    bias_lrelu_kernel<HID_C><<<(N_NODES * HID_C + 255) / 256, 256, 0, stream>>>(a1, b1, N_NODES * HID_C);

    // layer 2: h2 = a1 @ W2 ; scatter ; bias + leaky-relu
    gemm_wmma_kernel<HID_C, OUT_C><<<N_NODES / 32, 64, 0, stream>>>(a1, W2, h2);
    {
        long long total = (long long)N_EDGES * (OUT_C / 16);
        scatter_kernel<OUT_C><<<(int)((total + 255) / 256), 256, 0, stream>>>(rows, cols, dis, h2, a2);
    }
    bias_lrelu_kernel<OUT_C><<<(N_NODES * OUT_C + 255) / 256, 256, 0, stream>>>(a2, b2, N_NODES * OUT_C);

    // mean pool + head
    pool_kernel<<<(N_NODES + 511) / 512, 256, 0, stream>>>(a2, bat, sums, counts);
    head_kernel<<<1, 64, 0, stream>>>(sums, counts, Wm, bm, out);
}